// CPSFFusedCodebook_23441931501924
// MI455X (gfx1250) — compile-verified
//
#include <hip/hip_runtime.h>

// Problem sizes (fixed by the reference)
#define B_SZ 2048
#define M_SZ 4096
#define N_SZ 64
#define S_SZ 64
#define EPS_TOTAL 1e-3f
#define M_CHUNKS 4
#define M_CHUNK (M_SZ / M_CHUNKS)

typedef float v2f __attribute__((ext_vector_type(2)));
typedef float v8f __attribute__((ext_vector_type(8)));

// 8-point Gauss-Legendre on [0,1]: t = 0.5*(x+1), w = 0.5*w_leg
__device__ __constant__ float GL_T[8] = {
    0.019855071751231843f, 0.10166676129318664f, 0.2372337950418355f,
    0.40828267875217505f,  0.5917173212478249f,  0.7627662049581645f,
    0.8983332387068134f,   0.9801449282487682f};
__device__ __constant__ float GL_W[8] = {
    0.05061426814518813f, 0.11119051722668723f, 0.15685332293894365f,
    0.181341891689181f,   0.181341891689181f,   0.15685332293894365f,
    0.11119051722668723f, 0.05061426814518813f};

// Full-precision f32 WMMA, 16x16 tile, K=4 per instruction.
__device__ __forceinline__ v8f wmma_f32(v2f a, v2f b, v8f c) {
  return __builtin_amdgcn_wmma_f32_16x16x4_f32(false, a, false, b, (short)0, c,
                                               false, false);
}

__device__ __forceinline__ float fast_exp(float x) {
  return __builtin_amdgcn_exp2f(x * 1.44269504088896340736f);  // v_exp_f32
}

// ---------------------------------------------------------------------------
// Kernel 0a: per-m and per-b scalar reductions over N=64.
// mpar layout: [0]=dd2, [1]=|zj|^2, [2]=c0r, [3]=c0i, [4]=1/sp2, [5]=1/so2
// bpar layout: [0..B)=|z|^2, [B..2B)=|d|^2
// ---------------------------------------------------------------------------
__global__ __launch_bounds__(256) void precomp_kernel(
    const float* __restrict__ z_re, const float* __restrict__ z_im,
    const float* __restrict__ d_re, const float* __restrict__ d_im,
    const float* __restrict__ zj_re, const float* __restrict__ zj_im,
    const float* __restrict__ dj_re, const float* __restrict__ dj_im,
    const float* __restrict__ sig_par, const float* __restrict__ sig_perp,
    float* __restrict__ mpar, float* __restrict__ bpar) {
  int i = blockIdx.x * blockDim.x + threadIdx.x;
  if (i < M_SZ) {
    const float* djr = dj_re + (size_t)i * N_SZ;
    const float* dji = dj_im + (size_t)i * N_SZ;
    const float* zjr = zj_re + (size_t)i * N_SZ;
    const float* zji = zj_im + (size_t)i * N_SZ;
    float dd2 = 0.f, zj2 = 0.f, c0r = 0.f, c0i = 0.f;
    for (int n = 0; n < N_SZ; ++n) {
      float dr = djr[n], di = dji[n], zr = zjr[n], zi = zji[n];
      dd2 += dr * dr + di * di;
      zj2 += zr * zr + zi * zi;
      c0r += dr * zr + di * zi;  // Re(conj(dj)*zj)
      c0i += dr * zi - di * zr;  // Im(conj(dj)*zj)
    }
    mpar[0 * M_SZ + i] = dd2;
    mpar[1 * M_SZ + i] = zj2;
    mpar[2 * M_SZ + i] = c0r;
    mpar[3 * M_SZ + i] = c0i;
    float sp = sig_par[i], so = sig_perp[i];
    mpar[4 * M_SZ + i] = 1.0f / (sp * sp + EPS_TOTAL);
    mpar[5 * M_SZ + i] = 1.0f / (so * so + EPS_TOTAL);
  }
  if (i < B_SZ) {
    const float* zr = z_re + (size_t)i * N_SZ;
    const float* zi = z_im + (size_t)i * N_SZ;
    const float* dr = d_re + (size_t)i * N_SZ;
    const float* di = d_im + (size_t)i * N_SZ;
    float z2 = 0.f, dsq = 0.f;
    for (int n = 0; n < N_SZ; ++n) {
      z2 += zr[n] * zr[n] + zi[n] * zi[n];
      dsq += dr[n] * dr[n] + di[n] * di[n];
    }
    bpar[i] = z2;
    bpar[B_SZ + i] = dsq;
  }
}

// ---------------------------------------------------------------------------
// Kernel 0b: transpose That (M,S) -> (S,M) so the output GEMM's B operand is
// a contiguous float2 per lane in the K (m) direction.
// ---------------------------------------------------------------------------
__global__ __launch_bounds__(256) void transpose_that_kernel(
    const float* __restrict__ that_re, const float* __restrict__ that_im,
    float* __restrict__ tT_re, float* __restrict__ tT_im) {
  int idx = blockIdx.x * blockDim.x + threadIdx.x;  // 0 .. S*M-1
  int m = idx & (M_SZ - 1);
  int s = idx >> 12;  // M_SZ = 2^12
  tT_re[idx] = that_re[(size_t)m * S_SZ + s];
  tT_im[idx] = that_im[(size_t)m * S_SZ + s];
}

// ---------------------------------------------------------------------------
// Kernel 1: one wave per 16(b) x 16(m) tile.
// Phase A: 4 accumulators (ipr, ipi, rzj, a) via chained f32 WMMA over N=64.
// Phase B: 8-node quadrature + angular term in the C/D register layout,
//          heavy on v_exp_f32; writes real wgt[b,m] tile.
// ---------------------------------------------------------------------------
__global__ __launch_bounds__(128) void wgt_kernel(
    const float* __restrict__ z_re, const float* __restrict__ z_im,
    const float* __restrict__ d_re, const float* __restrict__ d_im,
    const float* __restrict__ zj_re, const float* __restrict__ zj_im,
    const float* __restrict__ dj_re, const float* __restrict__ dj_im,
    const float* __restrict__ alpha, const float* __restrict__ mpar,
    const float* __restrict__ bpar, float* __restrict__ wgt) {
  const int lane = threadIdx.x & 31;
  const int wid = threadIdx.x >> 5;
  const int gw = blockIdx.x * 4 + wid;  // 32768 waves total
  const int bt = gw & 127;              // 128 b-tiles
  const int mt = gw >> 7;               // 256 m-tiles
  const int b0 = bt * 16, m0 = mt * 16;
  const int lm = lane & 15, half = lane >> 4;

  const size_t arow = (size_t)(b0 + lm) * N_SZ;  // A: row = b
  const size_t brow = (size_t)(m0 + lm) * N_SZ;  // B: col = m

  v8f ipr = {}, ipi = {}, rzj = {}, aacc = {};
#pragma unroll 4
  for (int k0 = 0; k0 < N_SZ; k0 += 4) {
    const int kk = k0 + 2 * half;
    v2f azr = *(const v2f*)(z_re + arow + kk);
    v2f azi = *(const v2f*)(z_im + arow + kk);
    v2f adr = *(const v2f*)(d_re + arow + kk);
    v2f adi = *(const v2f*)(d_im + arow + kk);
    v2f bdr = *(const v2f*)(dj_re + brow + kk);
    v2f bdi = *(const v2f*)(dj_im + brow + kk);
    v2f bzr = *(const v2f*)(zj_re + brow + kk);
    v2f bzi = *(const v2f*)(zj_im + brow + kk);
    v2f bndi = -bdi;  // f32 WMMA has no A/B NEG modifier

    ipr = wmma_f32(azr, bdr, ipr);    // + z_re . dj_re^T
    ipr = wmma_f32(azi, bdi, ipr);    // + z_im . dj_im^T
    ipi = wmma_f32(azi, bdr, ipi);    // + z_im . dj_re^T
    ipi = wmma_f32(azr, bndi, ipi);   // - z_re . dj_im^T
    rzj = wmma_f32(azr, bzr, rzj);    // + z_re . zj_re^T
    rzj = wmma_f32(azi, bzi, rzj);    // + z_im . zj_im^T
    aacc = wmma_f32(adr, bdr, aacc);  // + d_re . dj_re^T
    aacc = wmma_f32(adi, bdi, aacc);  // + d_im . dj_im^T
  }

  const int m = m0 + lm;
  const float dd2 = mpar[0 * M_SZ + m];
  const float zj2 = mpar[1 * M_SZ + m];
  const float c0r = mpar[2 * M_SZ + m];
  const float c0i = mpar[3 * M_SZ + m];
  const float isp2 = mpar[4 * M_SZ + m];
  const float iso2 = mpar[5 * M_SZ + m];
  const float al = alpha[m];

#pragma unroll
  for (int r = 0; r < 8; ++r) {
    const int b = b0 + r + 8 * half;
    const float z2 = bpar[b];
    const float dsq = bpar[B_SZ + b];
    const float eipr = ipr[r] - c0r;
    const float eipi = ipi[r] - c0i;
    const float q0 = z2 - 2.0f * rzj[r] + zj2;
    float eta = 0.0f;
#pragma unroll
    for (int k = 0; k < 8; ++k) {
      const float t = GL_T[k];
      const float wpr = eipr - t * dd2;
      const float qpar = wpr * wpr + eipi * eipi;
      const float q2 = q0 - 2.0f * t * eipr + t * t * dd2;
      const float qperp = fmaxf(q2 - qpar, 0.0f);
      eta += GL_W[k] * fast_exp(-0.5f * (qpar * isp2 + qperp * iso2));
    }
    eta *= al;
    const float dist = dsq - 2.0f * aacc[r] + dd2;
    const float ang = fast_exp(-0.5f * dist * iso2);
    wgt[(size_t)b * M_SZ + m] = eta * ang;
  }
}

// ---------------------------------------------------------------------------
// Kernel 2: partial T[b,s] = sum_{m in chunk} wgt[b,m] * That[m,s] (re, im).
// One wave per (16b x 16s output tile, m-chunk): 2048 waves -> real occupancy.
// 4 independent WMMA chains per wave; all operands contiguous float2 loads.
// Partials stored interleaved-complex, one copy per chunk.
// ---------------------------------------------------------------------------
__global__ __launch_bounds__(128) void out_gemm_kernel(
    const float* __restrict__ wgt, const float* __restrict__ tT_re,
    const float* __restrict__ tT_im, float* __restrict__ part) {
  const int lane = threadIdx.x & 31;
  const int wid = threadIdx.x >> 5;
  const int gw = blockIdx.x * 4 + wid;  // 2048 waves total
  const int bt = gw & 127;              // 128 b-tiles
  const int rest = gw >> 7;             // 0..15
  const int st = rest & 3;              // 4 s-tiles
  const int mc = rest >> 2;             // 4 m-chunks
  const int b0 = bt * 16, s0 = st * 16;
  const int mbase = mc * M_CHUNK;
  const int lm = lane & 15, half = lane >> 4;
  const int s = s0 + lm;

  const float* wrow = wgt + (size_t)(b0 + lm) * M_SZ + mbase;   // A: row = b
  const float* trow_r = tT_re + (size_t)s * M_SZ + mbase;       // B: col = s
  const float* trow_i = tT_im + (size_t)s * M_SZ + mbase;

  v8f tre0 = {}, tre1 = {}, tim0 = {}, tim1 = {};
#pragma unroll 4
  for (int m0 = 0; m0 < M_CHUNK; m0 += 8) {
    const int mk0 = m0 + 2 * half;
    const int mk1 = m0 + 4 + 2 * half;
    v2f aw0 = *(const v2f*)(wrow + mk0);
    v2f aw1 = *(const v2f*)(wrow + mk1);
    v2f br0 = *(const v2f*)(trow_r + mk0);
    v2f br1 = *(const v2f*)(trow_r + mk1);
    v2f bi0 = *(const v2f*)(trow_i + mk0);
    v2f bi1 = *(const v2f*)(trow_i + mk1);
    tre0 = wmma_f32(aw0, br0, tre0);
    tim0 = wmma_f32(aw0, bi0, tim0);
    tre1 = wmma_f32(aw1, br1, tre1);
    tim1 = wmma_f32(aw1, bi1, tim1);
  }
  const v8f tre = tre0 + tre1;
  const v8f tim = tim0 + tim1;

  float* pbase = part + (size_t)mc * (2 * B_SZ * S_SZ);
#pragma unroll
  for (int r = 0; r < 8; ++r) {
    const int b = b0 + r + 8 * half;
    v2f o = {tre[r], tim[r]};
    *(v2f*)(pbase + 2 * ((size_t)b * S_SZ + s)) = o;
  }
}

// ---------------------------------------------------------------------------
// Kernel 3: reduce the 4 m-chunk partials into d_out (interleaved complex64).
// ---------------------------------------------------------------------------
__global__ __launch_bounds__(256) void reduce_kernel(
    const float* __restrict__ part, float* __restrict__ out) {
  int j = blockIdx.x * blockDim.x + threadIdx.x;  // 0 .. 2*B*S-1
  const size_t stride = (size_t)2 * B_SZ * S_SZ;
  out[j] = part[j] + part[stride + j] + part[2 * stride + j] +
           part[3 * stride + j];
}

// ---------------------------------------------------------------------------
// Host launcher
// Inputs (dict order): 0 z_re, 1 z_im, 2 d_re, 3 d_im, 4 zj_re, 5 zj_im,
//                      6 dj_re, 7 dj_im, 8 That_re, 9 That_im,
//                      10 alpha, 11 sig_par, 12 sig_perp
// Workspace: wgt (B*M) | mpar (6M) | bpar (2B) | tT_re (S*M) | tT_im (S*M)
//            | part (4 * 2*B*S)                       (~38.3 MB total, f32)
// ---------------------------------------------------------------------------
extern "C" void kernel_launch(void* const* d_in, const int* in_sizes, int n_in,
                              void* d_out, int out_size, void* d_ws,
                              size_t ws_size, hipStream_t stream) {
  const float* z_re = (const float*)d_in[0];
  const float* z_im = (const float*)d_in[1];
  const float* d_re = (const float*)d_in[2];
  const float* d_im = (const float*)d_in[3];
  const float* zj_re = (const float*)d_in[4];
  const float* zj_im = (const float*)d_in[5];
  const float* dj_re = (const float*)d_in[6];
  const float* dj_im = (const float*)d_in[7];
  const float* that_re = (const float*)d_in[8];
  const float* that_im = (const float*)d_in[9];
  const float* alpha = (const float*)d_in[10];
  const float* sig_par = (const float*)d_in[11];
  const float* sig_perp = (const float*)d_in[12];

  float* ws = (float*)d_ws;
  float* wgt = ws;                              // B*M
  float* mpar = wgt + (size_t)B_SZ * M_SZ;      // 6*M
  float* bpar = mpar + 6 * (size_t)M_SZ;        // 2*B
  float* tT_re = bpar + 2 * (size_t)B_SZ;       // S*M
  float* tT_im = tT_re + (size_t)S_SZ * M_SZ;   // S*M
  float* part = tT_im + (size_t)S_SZ * M_SZ;    // 4 * 2*B*S

  precomp_kernel<<<16, 256, 0, stream>>>(z_re, z_im, d_re, d_im, zj_re, zj_im,
                                         dj_re, dj_im, sig_par, sig_perp, mpar,
                                         bpar);

  transpose_that_kernel<<<(S_SZ * M_SZ) / 256, 256, 0, stream>>>(
      that_re, that_im, tT_re, tT_im);

  // 32768 tiles (128 b-tiles x 256 m-tiles), 4 waves per 128-thread block
  wgt_kernel<<<8192, 128, 0, stream>>>(z_re, z_im, d_re, d_im, zj_re, zj_im,
                                       dj_re, dj_im, alpha, mpar, bpar, wgt);

  // 2048 waves (128 b-tiles x 4 s-tiles x 4 m-chunks)
  out_gemm_kernel<<<512, 128, 0, stream>>>(wgt, tT_re, tT_im, part);

  // 2*B*S = 262144 output floats
  reduce_kernel<<<(2 * B_SZ * S_SZ) / 256, 256, 0, stream>>>(part,
                                                             (float*)d_out);
}